// renderModule_87677462380876
// MI455X (gfx1250) — compile-verified
//
#include <hip/hip_runtime.h>
#include <stdint.h>

// ---------------- problem constants (from reference) ----------------
#define HH 1024
#define WW 1024
#define NPTS 4000000
#define EHH 512
#define EWW 1024
#define EPSV 1e-6f

// ---------------- tiling for async-LDS staged scatter ----------------
#define THREADS 256
#define TILE_PTS 512
#define TILE_BYTES (TILE_PTS * 24)                      // 12288 B
#define CHUNKS_PER_THREAD (TILE_BYTES / (THREADS * 16)) // 3 x b128 per thread

// ---------------- CDNA5 async global->LDS helpers ----------------
// The clang builtin takes (AS1 v4i* src, AS3 v4i* dst, imm offset, imm cpol).
// We materialize both pointers via inttoptr: an AS1 pointer's value is the
// flat global address; an AS3 pointer's value is the LDS byte offset (the
// flat address of a __shared__ object carries it in the low 32 bits).
typedef int v4i __attribute__((vector_size(16)));
typedef __attribute__((address_space(1))) v4i v4i_g;  // global
typedef __attribute__((address_space(3))) v4i v4i_l;  // LDS

__device__ __forceinline__ void async_ld_b128(unsigned lds_off, const void* gsrc) {
#if defined(__has_builtin) && __has_builtin(__builtin_amdgcn_global_load_async_to_lds_b128)
  __builtin_amdgcn_global_load_async_to_lds_b128(
      (v4i_g*)(uintptr_t)gsrc,
      (v4i_l*)(uintptr_t)lds_off,
      0, 0);
#else
  asm volatile("global_load_async_to_lds_b128 %0, %1, off"
               :: "v"(lds_off), "v"(gsrc) : "memory");
#endif
}

__device__ __forceinline__ void wait_async_le3() {
#if defined(__has_builtin) && __has_builtin(__builtin_amdgcn_s_wait_asynccnt)
  __builtin_amdgcn_s_wait_asynccnt(3);
#else
  asm volatile("s_wait_asynccnt 3" ::: "memory");
#endif
}

__device__ __forceinline__ void wait_async_le0() {
#if defined(__has_builtin) && __has_builtin(__builtin_amdgcn_s_wait_asynccnt)
  __builtin_amdgcn_s_wait_asynccnt(0);
#else
  asm volatile("s_wait_asynccnt 0" ::: "memory");
#endif
}

// ---------------- pass A: clear accumulators ----------------
__global__ void rm877_zero_ws(float4* __restrict__ ws, int n4) {
  int i = blockIdx.x * blockDim.x + threadIdx.x;
  if (i < n4) ws[i] = make_float4(0.f, 0.f, 0.f, 0.f);
}

// ---------------- pass B: async-staged point splat ----------------
__global__ __launch_bounds__(THREADS)
void rm877_splat(const float* __restrict__ cam, const float* __restrict__ pts,
                 float* __restrict__ acc, float* __restrict__ wsum, int nTiles) {
  __shared__ __align__(16) float lds[2][TILE_BYTES / 4];
  const int tid = threadIdx.x;

  const float fx = cam[0], fy = cam[1], cx = cam[2], cy = cam[3];
  const float r00 = cam[4],  r01 = cam[5],  r02 = cam[6];
  const float r10 = cam[7],  r11 = cam[8],  r12 = cam[9];
  const float r20 = cam[10], r21 = cam[11], r22 = cam[12];
  const float t0 = cam[13], t1 = cam[14], t2 = cam[15];

  const size_t maxOff = (size_t)NPTS * 24 - 16;  // 16B-aligned last chunk
  unsigned ldsBase[2];
  ldsBase[0] = (unsigned)(uintptr_t)&lds[0][0];
  ldsBase[1] = (unsigned)(uintptr_t)&lds[1][0];

  int tile = blockIdx.x;
  const int stride = (int)gridDim.x;

  if (tile < nTiles) {
    // prologue: stage first tile into buffer 0 (ASYNCcnt += 3 per wave)
#pragma unroll
    for (int c = 0; c < CHUNKS_PER_THREAD; ++c) {
      size_t off = (size_t)tile * TILE_BYTES + (size_t)(c * THREADS + tid) * 16;
      if (off > maxOff) off = maxOff;  // clamp keeps issue count exact on tail
      async_ld_b128(ldsBase[0] + (unsigned)(c * THREADS + tid) * 16,
                    (const char*)pts + off);
    }
  }

  int cur = 0;
  for (; tile < nTiles; tile += stride) {
    const int nxt = tile + stride;
    const bool hasNext = nxt < nTiles;
    if (hasNext) {
      // prefetch next tile into the other buffer, then wait only for current
#pragma unroll
      for (int c = 0; c < CHUNKS_PER_THREAD; ++c) {
        size_t off = (size_t)nxt * TILE_BYTES + (size_t)(c * THREADS + tid) * 16;
        if (off > maxOff) off = maxOff;
        async_ld_b128(ldsBase[cur ^ 1] + (unsigned)(c * THREADS + tid) * 16,
                      (const char*)pts + off);
      }
      wait_async_le3();   // <=3 outstanding: only the next-tile prefetch
    } else {
      wait_async_le0();
    }
    __syncthreads();      // all waves' current-tile DMA visible in LDS

    const float* buf = &lds[cur][0];
#pragma unroll
    for (int s = 0; s < TILE_PTS / THREADS; ++s) {
      const int p = s * THREADS + tid;
      const long gp = (long)tile * TILE_PTS + p;
      if (gp < (long)NPTS) {
        // 24B/point, 8B-aligned: 3x b64 LDS reads, conflict-free banks
        float2 a = *(const float2*)(buf + p * 6 + 0);
        float2 b = *(const float2*)(buf + p * 6 + 2);
        float2 c2 = *(const float2*)(buf + p * 6 + 4);
        const float px = a.x, py = a.y, pz = b.x;
        const float cr = b.y, cg = c2.x, cb = c2.y;

        const float pcx = fmaf(r00, px, fmaf(r01, py, fmaf(r02, pz, t0)));
        const float pcy = fmaf(r10, px, fmaf(r11, py, fmaf(r12, pz, t1)));
        const float pcz = fmaf(r20, px, fmaf(r21, py, fmaf(r22, pz, t2)));

        const float zs = fmaxf(pcz, EPSV);
        const float u = fx * pcx / zs + cx;
        const float v = fy * pcy / zs + cy;
        const float fu = floorf(u), fv = floorf(v);
        const bool valid = (pcz > EPSV) & (fu >= 0.f) & (fu < (float)WW) &
                           (fv >= 0.f) & (fv < (float)HH);
        if (valid) {
          const int idx = (int)fv * WW + (int)fu;
          const float wgt = 1.0f / zs;
          float* ap = acc + (size_t)idx * 4;
          atomicAdd(ap + 0, wgt * cr);
          atomicAdd(ap + 1, wgt * cg);
          atomicAdd(ap + 2, wgt * cb);
          atomicAdd(ap + 3, wgt * pcz);
          atomicAdd(wsum + idx, wgt);
        }
      }
    }
    __syncthreads();  // everyone done reading buf[cur] before it is re-filled
    cur ^= 1;
  }
}

// ---------------- pass C: resolve / environment background ----------------
__global__ __launch_bounds__(256)
void rm877_resolve(const float* __restrict__ cam, const float* __restrict__ env,
                   const float* __restrict__ acc, const float* __restrict__ wsum,
                   float4* __restrict__ out) {
  const int pix = blockIdx.x * blockDim.x + threadIdx.x;
  if (pix >= HH * WW) return;
  const int y = pix >> 10;
  const int x = pix & (WW - 1);

  const float fx = cam[0], fy = cam[1], cx = cam[2], cy = cam[3];
  const float r00 = cam[4],  r01 = cam[5],  r02 = cam[6];
  const float r10 = cam[7],  r11 = cam[8],  r12 = cam[9];
  const float r20 = cam[10], r21 = cam[11], r22 = cam[12];

  const float dx = ((float)x + 0.5f - cx) / fx;
  const float dy = ((float)y + 0.5f - cy) / fy;
  // dirs = [dx, dy, 1] @ R
  float d0 = dx * r00 + dy * r10 + r20;
  float d1 = dx * r01 + dy * r11 + r21;
  float d2 = dx * r02 + dy * r12 + r22;
  const float inv = 1.0f / sqrtf(d0 * d0 + d1 * d1 + d2 * d2);
  d0 *= inv; d1 *= inv; d2 *= inv;

  const float theta = atan2f(d0, d2);
  const float phi = asinf(fminf(fmaxf(d1, -1.f), 1.f));
  int eu = (int)((theta * 0.15915494309189535f + 0.5f) * (float)EWW);
  eu = min(max(eu, 0), EWW - 1);
  int ev = (int)((phi * 0.3183098861837907f + 0.5f) * (float)EHH);
  ev = min(max(ev, 0), EHH - 1);

  float4 o;
  const float ws = wsum[pix];
  if (ws > 0.f) {
    const float inw = 1.0f / fmaxf(ws, EPSV);
    const float* a = acc + (size_t)pix * 4;
    o = make_float4(a[0] * inw, a[1] * inw, a[2] * inw, a[3] * inw);
  } else {
    const float* e = env + ((size_t)ev * EWW + eu) * 3;
    o = make_float4(e[0], e[1], e[2], 0.f);
  }
  out[pix] = o;
}

// ---------------- entry point ----------------
extern "C" void kernel_launch(void* const* d_in, const int* in_sizes, int n_in,
                              void* d_out, int out_size, void* d_ws, size_t ws_size,
                              hipStream_t stream) {
  const float* cam = (const float*)d_in[0];   // 16 floats
  const float* pts = (const float*)d_in[1];   // NPTS x 6
  const float* env = (const float*)d_in[2];   // EH x EW x 3
  // d_in[3], d_in[4]: cam_type / environment_type, always 0 here.

  float* acc  = (float*)d_ws;                 // H*W*4 f32
  float* wsum = acc + (size_t)HH * WW * 4;    // H*W   f32

  const int n4 = (HH * WW * 5) / 4;           // accumulators as float4
  rm877_zero_ws<<<(n4 + 255) / 256, 256, 0, stream>>>((float4*)d_ws, n4);

  const int nTiles = (NPTS + TILE_PTS - 1) / TILE_PTS;
  rm877_splat<<<2048, THREADS, 0, stream>>>(cam, pts, acc, wsum, nTiles);

  rm877_resolve<<<(HH * WW + 255) / 256, 256, 0, stream>>>(
      cam, env, acc, wsum, (float4*)d_out);
}